// Critic_GCN_69922067579337
// MI455X (gfx1250) — compile-verified
//
#include <hip/hip_runtime.h>

#define BB 4
#define NN 10000
#define EE 320000
#define FF 128
#define HH 128
#define PAD 4         // LDS row padding (floats): 132 mod 64 = 4 -> conflict-free b64 pairs

typedef float v2f __attribute__((ext_vector_type(2)));
typedef float v8f __attribute__((ext_vector_type(8)));
typedef unsigned int uint32x4 __attribute__((ext_vector_type(4)));
typedef int int32x4 __attribute__((ext_vector_type(4)));
typedef int int32x8 __attribute__((ext_vector_type(8)));

#if defined(__has_builtin)
#if __has_builtin(__builtin_amdgcn_tensor_load_to_lds) && \
    __has_builtin(__builtin_amdgcn_s_wait_tensorcnt)
#define HAVE_TDM 1
#endif
#endif

// ---------------- degree / norm precompute ----------------

__global__ void k_zero(float* __restrict__ p, int n) {
    int i = blockIdx.x * blockDim.x + threadIdx.x;
    if (i < n) p[i] = 0.0f;
}

__global__ void k_deg(const int* __restrict__ edges, const float* __restrict__ attr,
                      float* __restrict__ deg) {
    int i = blockIdx.x * blockDim.x + threadIdx.x;
    if (i >= BB * EE) return;
    int b = i / EE, e = i - b * EE;
    int col = edges[b * 2 * EE + EE + e];
    unsafeAtomicAdd(&deg[b * NN + col], attr[b * EE + e]);
}

__global__ void k_dinv(float* __restrict__ deg, int n) {
    int i = blockIdx.x * blockDim.x + threadIdx.x;
    if (i >= n) return;
    float d = deg[i];
    deg[i] = (d > 0.0f) ? rsqrtf(d) : 0.0f;
}

__global__ void k_norm(const int* __restrict__ edges, const float* __restrict__ attr,
                       const float* __restrict__ dinv, float* __restrict__ nrm) {
    int i = blockIdx.x * blockDim.x + threadIdx.x;
    if (i >= BB * EE) return;
    int b = i / EE, e = i - b * EE;
    const int* eb = edges + b * 2 * EE;
    int row = eb[e];
    int col = eb[EE + e];
    nrm[b * EE + e] = dinv[b * NN + row] * attr[b * EE + e] * dinv[b * NN + col];
}

// ---------------- dense GEMM  Out[b] = (relu?)A[b] @ W  via f32 WMMA ----------------
// grid = (NN/16, BB), block = 256 (8 waves). wave w -> cols [16w,16w+16).
// A tile (16 rows x 128 k, 8 KB contiguous) staged to LDS via the Tensor Data Mover
// (pad_enable inserts the 4-dword row padding); W staged transposed by all threads
// while the TDM runs. A-matrix 16x4 f32 layout: lanes 0-15 hold K=k0+0/k0+1,
// lanes 16-31 hold K=k0+2/k0+3. B/C/D rows striped across lanes.

template <int RELU>
__global__ void __launch_bounds__(256)
k_gemm(const float* __restrict__ A, const float* __restrict__ W,
       float* __restrict__ Out) {
    __shared__ float Wt[HH * (FF + PAD)];   // W transposed: Wt[n][k]
    __shared__ float At[16 * (FF + PAD)];   // A tile row-major: At[m][k]
    const int b  = blockIdx.y;
    const int m0 = blockIdx.x * 16;
    const float* Ab = A   + (size_t)b * NN * FF;
    float*       Ob = Out + (size_t)b * NN * HH;

#if HAVE_TDM
    if (threadIdx.x == 0) {
        // TDM D#: 1D copy of 1024 x 8B = 8 KB, LDS padding 4 dwords every 128 dwords.
        unsigned long long ga = (unsigned long long)(uintptr_t)(Ab + (size_t)m0 * FF);
        unsigned int la = (unsigned int)(uintptr_t)
            (__attribute__((address_space(3))) float*)At;
        uint32x4 g0;
        g0.x = 1u;                                            // count=1 (valid user D#)
        g0.y = la;                                            // lds_addr
        g0.z = (unsigned int)(ga & 0xFFFFFFFFu);              // global_addr[31:0]
        g0.w = (unsigned int)((ga >> 32) & 0x01FFFFFFu) | (2u << 30);  // [56:32] | type=2
        int32x8 g1;
        g1[0] = (3 << 16)        // data_size = 8B
              | (1 << 20)        // pad_enable
              | (6 << 22)        // pad_interval: 128 dwords
              | (3 << 25);       // pad_amount: 4 dwords
        g1[1] = (int)(1024u << 16);   // tensor_dim0[15:0] = 1024 (bits 63:48)
        g1[2] = (int)(1u << 16);      // tensor_dim0 hi = 0; tensor_dim1 = 1
        g1[3] = (int)(1024u << 16);   // tensor_dim1 hi = 0; tile_dim0 = 1024
        g1[4] = 0;                    // tile_dim1 = 0 (unused), tile_dim2 = 0
        g1[5] = 1024;                 // tensor_dim0_stride[31:0]
        g1[6] = 0;
        g1[7] = 0;
        int32x4 gz = {0, 0, 0, 0};
#if __clang_major__ >= 23
        int32x8 gz8 = {0, 0, 0, 0, 0, 0, 0, 0};
        __builtin_amdgcn_tensor_load_to_lds(g0, g1, gz, gz, gz8, 0);
#else
        __builtin_amdgcn_tensor_load_to_lds(g0, g1, gz, gz, 0);
#endif
    }
#else
    for (int i = threadIdx.x; i < 16 * (FF / 4); i += 256) {  // 512 float4 chunks
        int r  = i >> 5;
        int cq = i & 31;
        *(float4*)(At + r * (FF + PAD) + cq * 4) =
            *(const float4*)(Ab + (size_t)(m0 + r) * FF + cq * 4);
    }
#endif

    // stage W transposed (overlaps with the TDM transfer)
    for (int i = threadIdx.x; i < FF * HH; i += 256) {
        int k = i >> 7;                      // W is [F][H] row-major
        int n = i & 127;
        Wt[n * (FF + PAD) + k] = W[i];
    }

#if HAVE_TDM
    if (threadIdx.x == 0) __builtin_amdgcn_s_wait_tensorcnt(0);
#endif
    __syncthreads();

    const int wave   = threadIdx.x >> 5;
    const int lane   = threadIdx.x & 31;
    const int n0     = wave * 16;
    const int laneM  = lane & 15;
    const int hiHalf = lane >> 4;            // 0: lanes 0-15, 1: lanes 16-31
    const int koff   = hiHalf * 2;

    const float* arow = At + (size_t)laneM * (FF + PAD);
    const float* brow = Wt + (size_t)(n0 + laneM) * (FF + PAD);

    v8f c = {};
#pragma unroll
    for (int k0 = 0; k0 < FF; k0 += 4) {
        v2f a, bm;
        a.x = arow[k0 + koff];
        a.y = arow[k0 + koff + 1];
        if (RELU) { a.x = fmaxf(a.x, 0.0f); a.y = fmaxf(a.y, 0.0f); }
        bm.x = brow[k0 + koff];
        bm.y = brow[k0 + koff + 1];
        c = __builtin_amdgcn_wmma_f32_16x16x4_f32(false, a, false, bm,
                                                  (short)0, c, false, false);
    }

    const int mS = m0 + hiHalf * 8;          // C/D: VGPR i -> M = i (+8 for hi lanes)
#pragma unroll
    for (int i = 0; i < 8; ++i)
        Ob[(size_t)(mS + i) * HH + n0 + laneM] = c[i];
}

// ---------------- bias init + edge scatter ----------------

__global__ void k_initb(float* __restrict__ h, const float* __restrict__ bias) {
    int i = blockIdx.x * blockDim.x + threadIdx.x;
    if (i < BB * NN * HH) h[i] = bias[i & (HH - 1)];
}

__global__ void k_scatter(const int* __restrict__ edges, const float* __restrict__ nrm,
                          const float* __restrict__ xw, float* __restrict__ out) {
    int i = blockIdx.x * blockDim.x + threadIdx.x;
    if (i >= BB * EE * (HH / 4)) return;
    int b  = i / (EE * 32);
    int r  = i - b * (EE * 32);
    int e  = r >> 5;                          // one edge per wave: lanes cover 128 floats
    int h4 = (r & 31) << 2;
    const int* eb = edges + b * 2 * EE;
    int row = eb[e];
    int col = eb[EE + e];
    float nm = nrm[b * EE + e];
    const float4 v = *(const float4*)(xw + ((size_t)b * NN + row) * HH + h4);
    float* o = out + ((size_t)b * NN + col) * HH + h4;
    unsafeAtomicAdd(o + 0, nm * v.x);
    unsafeAtomicAdd(o + 1, nm * v.y);
    unsafeAtomicAdd(o + 2, nm * v.z);
    unsafeAtomicAdd(o + 3, nm * v.w);
}

// ---------------- head: mean(relu(h3)) @ Wl + bl ----------------

__global__ void __launch_bounds__(256)
k_head(const float* __restrict__ h, const float* __restrict__ Wl,
       const float* __restrict__ bl, float* __restrict__ out) {
    __shared__ float s[256];
    const int b = blockIdx.x;
    const int t = threadIdx.x;
    const int hh   = t & 127;
    const int half = t >> 7;
    const float* hb = h + (size_t)b * NN * HH;
    float acc = 0.0f;
    for (int n = half; n < NN; n += 2)
        acc += fmaxf(hb[(size_t)n * HH + hh], 0.0f);
    s[t] = acc;
    __syncthreads();
    if (t < 128) s[t] = (s[t] + s[t + 128]) * Wl[t];
    __syncthreads();
    for (int st = 64; st > 0; st >>= 1) {
        if (t < st) s[t] += s[t + st];
        __syncthreads();
    }
    if (t == 0) out[b] = s[0] / (float)NN + bl[0];
}

// ---------------- launcher ----------------

extern "C" void kernel_launch(void* const* d_in, const int* in_sizes, int n_in,
                              void* d_out, int out_size, void* d_ws, size_t ws_size,
                              hipStream_t stream) {
    const float* feat  = (const float*)d_in[0];   // [B,N,F]
    const int*   edges = (const int*)  d_in[1];   // [B,2,E]
    const float* attr  = (const float*)d_in[2];   // [B,E]
    const float* W1    = (const float*)d_in[3];
    const float* b1    = (const float*)d_in[4];
    const float* W2    = (const float*)d_in[5];
    const float* b2    = (const float*)d_in[6];
    const float* W3    = (const float*)d_in[7];
    const float* b3    = (const float*)d_in[8];
    const float* Wl    = (const float*)d_in[9];
    const float* bl    = (const float*)d_in[10];
    float* out = (float*)d_out;

    float* deg  = (float*)d_ws;                   // B*N
    float* nrm  = deg + BB * NN;                  // B*E
    float* xw   = nrm + BB * EE;                  // B*N*H
    float* h0   = xw  + (size_t)BB * NN * HH;     // B*N*H
    float* h1   = h0  + (size_t)BB * NN * HH;     // B*N*H

    const int T = 256;
    dim3 gemmGrid(NN / 16, BB);
    int nBN   = BB * NN;
    int nBE   = BB * EE;
    int nBNH  = BB * NN * HH;
    int nScat = BB * EE * (HH / 4);

    // edge norms (shared by all 3 layers)
    k_zero<<<(nBN + T - 1) / T, T, 0, stream>>>(deg, nBN);
    k_deg <<<(nBE + T - 1) / T, T, 0, stream>>>(edges, attr, deg);
    k_dinv<<<(nBN + T - 1) / T, T, 0, stream>>>(deg, nBN);
    k_norm<<<(nBE + T - 1) / T, T, 0, stream>>>(edges, attr, deg, nrm);

    // layer 1: x @ W1 -> scatter into h0 (init b1)
    k_gemm<0><<<gemmGrid, T, 0, stream>>>(feat, W1, xw);
    k_initb  <<<(nBNH + T - 1) / T, T, 0, stream>>>(h0, b1);
    k_scatter<<<(nScat + T - 1) / T, T, 0, stream>>>(edges, nrm, xw, h0);

    // layer 2: relu(h0) @ W2 -> scatter into h1 (init b2)
    k_gemm<1><<<gemmGrid, T, 0, stream>>>(h0, W2, xw);
    k_initb  <<<(nBNH + T - 1) / T, T, 0, stream>>>(h1, b2);
    k_scatter<<<(nScat + T - 1) / T, T, 0, stream>>>(edges, nrm, xw, h1);

    // layer 3: relu(h1) @ W3 -> scatter into h0 (init b3)
    k_gemm<1><<<gemmGrid, T, 0, stream>>>(h1, W3, xw);
    k_initb  <<<(nBNH + T - 1) / T, T, 0, stream>>>(h0, b3);
    k_scatter<<<(nScat + T - 1) / T, T, 0, stream>>>(edges, nrm, xw, h0);

    // head: mean(relu(h0)) @ Wl + bl
    k_head<<<BB, T, 0, stream>>>(h0, Wl, bl, out);
}